// GATGCN_72834055405953
// MI455X (gfx1250) — compile-verified
//
#include <hip/hip_runtime.h>
#include <hip/hip_bf16.h>
#include <math.h>

typedef __attribute__((ext_vector_type(2))) float v2f;
typedef __attribute__((ext_vector_type(8))) float v8f;

#define DFEAT 64
#define NEG_SLOPE 0.2f

// Float atomic-max via sign-magnitude int ordering trick (works for mixed signs,
// init value -inf). Lowers to global_atomic_max_i32 / global_atomic_min_u32.
__device__ __forceinline__ void atomicMaxF(float* addr, float v) {
    if (v >= 0.0f) {
        atomicMax((int*)addr, __float_as_int(v));
    } else {
        atomicMin((unsigned int*)addr, __float_as_uint(v));
    }
}

// ---------------------------------------------------------------------------
// K0: out[n][c] = bias[c]; m = -inf; s = 0
// ---------------------------------------------------------------------------
__global__ __launch_bounds__(256) void gat_init(float* __restrict__ out,
                                                const float* __restrict__ bias,
                                                float* __restrict__ mmax,
                                                float* __restrict__ ssum,
                                                int nnodes) {
    int i = blockIdx.x * blockDim.x + threadIdx.x;
    if (i < nnodes * DFEAT) out[i] = bias[i & (DFEAT - 1)];
    if (i < nnodes) {
        mmax[i] = __int_as_float(0xFF800000);  // -inf
        ssum[i] = 0.0f;
    }
}

// ---------------------------------------------------------------------------
// K1: xl = x @ W_l ; xr = x @ W_r   via V_WMMA_F32_16X16X4_F32 (wave32)
// Block = 8 waves: wave[2] selects {W_l, W_r}; wave[1:0] selects 16-col tile.
// Each wave owns one 16x16 output tile; K=64 covered by 16 WMMAs of K=4.
// ---------------------------------------------------------------------------
__global__ __launch_bounds__(256) void gat_gemm(const float* __restrict__ x,
                                                const float* __restrict__ Wl,
                                                const float* __restrict__ Wr,
                                                float* __restrict__ xl,
                                                float* __restrict__ xr,
                                                int nrows) {
    const int wave = threadIdx.x >> 5;
    const int lane = threadIdx.x & 31;
    const int m0   = blockIdx.x * 16;
    const float* __restrict__ W  = (wave & 4) ? Wr : Wl;
    float* __restrict__ out      = (wave & 4) ? xr : xl;
    const int n0 = (wave & 3) * 16;
    const bool hi = lane >= 16;     // high half-wave covers K+2/K+3 (A) etc.
    const int  ml = lane & 15;      // A: row (M); B/D: column (N)

    // clamp row reads for a possible ragged last tile (exact for N=50000)
    const int mrow = min(m0 + ml, nrows - 1);

    v8f acc = {};
    for (int kb = 0; kb < 16; ++kb) {
        const int k0 = kb * 4 + (hi ? 2 : 0);
        v2f a, b;
        const float* xp = x + (size_t)mrow * DFEAT + k0;
        a.x = xp[0];
        a.y = xp[1];
        const float* wp = W + (size_t)k0 * DFEAT + n0 + ml;
        b.x = wp[0];
        b.y = wp[DFEAT];
        acc = __builtin_amdgcn_wmma_f32_16x16x4_f32(
            /*neg_a=*/false, a, /*neg_b=*/false, b,
            /*c_mod=*/(short)0, acc, /*reuse_a=*/false, /*reuse_b=*/false);
    }

    const int col = n0 + ml;
    #pragma unroll
    for (int g = 0; g < 8; ++g) {
        const int row = m0 + g + (hi ? 8 : 0);   // D layout: VGPR g -> M=g / g+8
        if (row < nrows) out[(size_t)row * DFEAT + col] = acc[g];
    }
}

// ---------------------------------------------------------------------------
// K2: one wave per edge. Coalesced float2 gathers of xl[src], xr[dst] (L2-hot),
// leaky-ReLU, dot with att, wave32 shuffle reduction, atomic segment-max.
// ---------------------------------------------------------------------------
__global__ __launch_bounds__(256) void gat_edge_score(
    const long long* __restrict__ src, const long long* __restrict__ dst,
    const float* __restrict__ xl, const float* __restrict__ xr,
    const float* __restrict__ att, float* __restrict__ escore,
    float* __restrict__ mmax, int nedges) {
    const int e = blockIdx.x * 8 + (threadIdx.x >> 5);
    if (e >= nedges) return;
    const int lane = threadIdx.x & 31;
    const int s = (int)src[e];
    const int d = (int)dst[e];
    const float2 a = ((const float2*)(xl + (size_t)s * DFEAT))[lane];
    const float2 b = ((const float2*)(xr + (size_t)d * DFEAT))[lane];
    const float2 t = ((const float2*)att)[lane];
    float h0 = a.x + b.x; h0 = (h0 >= 0.0f) ? h0 : NEG_SLOPE * h0;
    float h1 = a.y + b.y; h1 = (h1 >= 0.0f) ? h1 : NEG_SLOPE * h1;
    float v = h0 * t.x + h1 * t.y;
    #pragma unroll
    for (int off = 16; off > 0; off >>= 1) v += __shfl_xor(v, off, 32);
    if (lane == 0) {
        escore[e] = v;
        atomicMaxF(&mmax[d], v);
    }
}

// ---------------------------------------------------------------------------
// K3: ex = exp(e - m[dst]) (in place); atomic segment-sum into ssum[dst].
// ---------------------------------------------------------------------------
__global__ __launch_bounds__(256) void gat_edge_exp(
    const long long* __restrict__ dst, float* __restrict__ escore,
    const float* __restrict__ mmax, float* __restrict__ ssum, int nedges) {
    const int i = blockIdx.x * blockDim.x + threadIdx.x;
    if (i >= nedges) return;
    const int d = (int)dst[i];
    const float ex = expf(escore[i] - mmax[d]);
    escore[i] = ex;
    atomicAdd(&ssum[d], ex);
}

// ---------------------------------------------------------------------------
// K4: 64 consecutive threads per edge: out[dst][c] += alpha * xl[src][c].
// Gather and atomic target both L2-resident (12.8 MB each).
// ---------------------------------------------------------------------------
__global__ __launch_bounds__(256) void gat_aggregate(
    const long long* __restrict__ src, const long long* __restrict__ dst,
    const float* __restrict__ xl, const float* __restrict__ escore,
    const float* __restrict__ ssum, float* __restrict__ out, int nedges) {
    const long long t = (long long)blockIdx.x * blockDim.x + threadIdx.x;
    const int i = (int)(t >> 6);
    const int c = (int)(t & 63);
    if (i >= nedges) return;
    const int s = (int)src[i];
    const int d = (int)dst[i];
    const float alpha = escore[i] / ssum[d];
    atomicAdd(&out[(size_t)d * DFEAT + c], alpha * xl[(size_t)s * DFEAT + c]);
}

// ---------------------------------------------------------------------------
extern "C" void kernel_launch(void* const* d_in, const int* in_sizes, int n_in,
                              void* d_out, int out_size, void* d_ws, size_t ws_size,
                              hipStream_t stream) {
    const float*     x    = (const float*)d_in[0];
    const long long* eidx = (const long long*)d_in[1];   // int64 [2, E]
    const float*     Wl   = (const float*)d_in[2];
    const float*     Wr   = (const float*)d_in[3];
    const float*     att  = (const float*)d_in[4];
    const float*     bias = (const float*)d_in[5];

    const int nnodes = in_sizes[0] / DFEAT;   // 50000
    const int nedges = in_sizes[1] / 2;       // 1250000
    const long long* src = eidx;
    const long long* dst = eidx + nedges;

    // Workspace layout (~31 MB): xl | xr | e | m | s
    char* ws = (char*)d_ws;
    const size_t nodeBytes = (size_t)nnodes * DFEAT * sizeof(float);
    float* xl     = (float*)ws;
    float* xr     = (float*)(ws + nodeBytes);
    float* escore = (float*)(ws + 2 * nodeBytes);
    float* mmax   = (float*)(ws + 2 * nodeBytes + (size_t)nedges * sizeof(float));
    float* ssum   = mmax + nnodes;
    float* out    = (float*)d_out;

    {
        const int total = nnodes * DFEAT;
        gat_init<<<(total + 255) / 256, 256, 0, stream>>>(out, bias, mmax, ssum, nnodes);
    }
    gat_gemm<<<(nnodes + 15) / 16, 256, 0, stream>>>(x, Wl, Wr, xl, xr, nnodes);
    gat_edge_score<<<(nedges + 7) / 8, 256, 0, stream>>>(src, dst, xl, xr, att,
                                                         escore, mmax, nedges);
    gat_edge_exp<<<(nedges + 255) / 256, 256, 0, stream>>>(dst, escore, mmax, ssum, nedges);
    {
        const long long tot = (long long)nedges * DFEAT;
        const int blocks = (int)((tot + 255) / 256);
        gat_aggregate<<<blocks, 256, 0, stream>>>(src, dst, xl, escore, ssum, out, nedges);
    }
}